// UDCP_22840636080960
// MI455X (gfx1250) — compile-verified
//
#include <hip/hip_runtime.h>
#include <hip/hip_bf16.h>

// ---------------- problem constants ----------------
static constexpr int   W     = 2048;
static constexpr int   H     = 2048;
static constexpr int   HW    = W * H;
static constexpr int   RMIN  = 7;      // PATCH=15 -> radius 7
static constexpr int   GF_R  = 60;     // guided filter radius (window 121)
static constexpr float OMEGA = 0.95f;
static constexpr float T0c   = 0.1f;
static constexpr float GFEPS = 1e-4f;
static constexpr int   K_TOP = 4194;   // int(HW * 0.001)

typedef __attribute__((ext_vector_type(2))) float v2f;
typedef __attribute__((ext_vector_type(8))) float v8f;

// CDNA5 feature probes (device pass only; host pass always sees same kernel symbols)
#if defined(__HIP_DEVICE_COMPILE__) && defined(__gfx1250__) && \
    __has_builtin(__builtin_amdgcn_wmma_f32_16x16x4_f32)
#define UDCP_WMMA 1
#else
#define UDCP_WMMA 0
#endif

#if defined(__HIP_DEVICE_COMPILE__) && defined(__gfx1250__) && \
    __has_builtin(__builtin_amdgcn_global_load_async_to_lds_b32) && \
    __has_builtin(__builtin_amdgcn_s_wait_asynccnt)
#define UDCP_ASYNC 1
#else
#define UDCP_ASYNC 0
#endif

__device__ __forceinline__ float boxcnt(int u) {
    int lo = u - GF_R; if (lo < 0) lo = 0;
    int hi = u + GF_R; if (hi > W - 1) hi = W - 1;
    return (float)(hi - lo + 1);
}

// ---------------- stats / histogram utilities ----------------
__global__ __launch_bounds__(256) void zero_stats_k(unsigned* stats) {
    int i = blockIdx.x * 256 + threadIdx.x;
    if (i < 8208) stats[i] = 0u;
}

// guide = 0.299R + 0.587G + 0.114B ; mgb = min(G,B)
__global__ __launch_bounds__(256) void prep_k(const float* __restrict__ img,
                                              float* __restrict__ guide,
                                              float* __restrict__ mgb) {
    size_t i = (size_t)blockIdx.x * 256 + threadIdx.x;
    float r = img[i], g = img[(size_t)HW + i], b = img[(size_t)2 * HW + i];
    guide[i] = 0.299f * r + 0.587f * g + 0.114f * b;
    mgb[i]   = fminf(g, b);
}

// horizontal 15-tap min filter; LDS-staged (async-to-LDS on CDNA5)
__global__ __launch_bounds__(256) void hmin_k(const float* __restrict__ src,
                                              float* __restrict__ dst) {
    __shared__ float sh[256 + 2 * RMIN];
    const int row = blockIdx.y;
    const int xs  = blockIdx.x * 256;
    const int t   = threadIdx.x;
    const float* srow = src + (size_t)row * W;
#if UDCP_ASYNC
    for (int i = t; i < 256 + 2 * RMIN; i += 256) {
        int gx = xs - RMIN + i;
        if (gx >= 0 && gx < W) {
            __builtin_amdgcn_global_load_async_to_lds_b32(
                (__attribute__((address_space(1))) int*)(srow + gx),
                (__attribute__((address_space(3))) int*)(sh + i), 0, 0);
        } else {
            sh[i] = __builtin_huge_valf();
        }
    }
    __builtin_amdgcn_s_wait_asynccnt(0);
    __syncthreads();
#else
    for (int i = t; i < 256 + 2 * RMIN; i += 256) {
        int gx = xs - RMIN + i;
        sh[i] = (gx >= 0 && gx < W) ? srow[gx] : __builtin_huge_valf();
    }
    __syncthreads();
#endif
    float m = __builtin_huge_valf();
#pragma unroll
    for (int d = 0; d <= 2 * RMIN; ++d) m = fminf(m, sh[t + d]);
    dst[(size_t)row * W + xs + t] = m;
}

// vertical 15-tap min; mode 1 applies t = 1 - OMEGA*min
__global__ __launch_bounds__(256) void vmin_k(const float* __restrict__ src,
                                              float* __restrict__ dst, int mode) {
    const int x = blockIdx.x * 256 + threadIdx.x;
    const int y = blockIdx.y;
    int y0 = y - RMIN; if (y0 < 0) y0 = 0;
    int y1 = y + RMIN; if (y1 > H - 1) y1 = H - 1;
    float m = __builtin_huge_valf();
    for (int yy = y0; yy <= y1; ++yy) m = fminf(m, src[(size_t)yy * W + x]);
    dst[(size_t)y * W + x] = mode ? (1.0f - OMEGA * m) : m;
}

// float-radix histograms for top-k threshold (dc >= 0 so bit order == value order)
__global__ __launch_bounds__(256) void hist_hi_k(const float* __restrict__ dc,
                                                 unsigned* __restrict__ hist) {
    size_t i = (size_t)blockIdx.x * 256 + threadIdx.x;
    unsigned bits = __float_as_uint(dc[i]);
    atomicAdd(&hist[bits >> 20], 1u);
}

__global__ void scan_k(const unsigned* __restrict__ hist, unsigned* sel, int mode) {
    if (threadIdx.x != 0) return;
    unsigned need = mode ? sel[1] : (unsigned)K_TOP;
    if (need == 0) need = 1;
    unsigned cum = 0;
    int b = 4095;
    for (; b > 0; --b) {
        unsigned h = hist[b];
        if (cum + h >= need) break;
        cum += h;
    }
    if (mode == 0) { sel[0] = (unsigned)b; sel[1] = need - cum; }
    else           { sel[2] = (sel[0] << 20) | ((unsigned)b << 8); }
}

__global__ __launch_bounds__(256) void hist_lo_k(const float* __restrict__ dc,
                                                 const unsigned* __restrict__ sel,
                                                 unsigned* __restrict__ hist) {
    size_t i = (size_t)blockIdx.x * 256 + threadIdx.x;
    unsigned bits = __float_as_uint(dc[i]);
    if ((bits >> 20) == sel[0]) atomicAdd(&hist[(bits >> 8) & 0xFFFu], 1u);
}

// per-channel max over pixels above dark-channel threshold (atmospheric light A)
__global__ __launch_bounds__(256) void amax_k(const float* __restrict__ img,
                                              const float* __restrict__ dc,
                                              const unsigned* __restrict__ sel,
                                              unsigned* __restrict__ am) {
    size_t i = (size_t)blockIdx.x * 256 + threadIdx.x;
    if (__float_as_uint(dc[i]) >= sel[2]) {
        atomicMax(&am[0], __float_as_uint(img[i]));
        atomicMax(&am[1], __float_as_uint(img[(size_t)HW + i]));
        atomicMax(&am[2], __float_as_uint(img[(size_t)2 * HW + i]));
    }
}

// rt = min(G/Ag, B/Ab)
__global__ __launch_bounds__(256) void tprep_k(const float* __restrict__ img,
                                               const unsigned* __restrict__ am,
                                               float* __restrict__ rt) {
    size_t i = (size_t)blockIdx.x * 256 + threadIdx.x;
    float Ag = __uint_as_float(am[1]);
    float Ab = __uint_as_float(am[2]);
    rt[i] = fminf(img[(size_t)HW + i] / Ag, img[(size_t)2 * HW + i] / Ab);
}

// horizontal 121-wide box sum; one row per block staged in LDS; optional product input
__global__ __launch_bounds__(256) void hbox_k(const float* __restrict__ s1,
                                              const float* s2,
                                              float* __restrict__ dst) {
    __shared__ float sh[W];
    const int row = blockIdx.x;
    const float* r1 = s1 + (size_t)row * W;
    const float* r2 = s2 ? s2 + (size_t)row * W : nullptr;
    for (int i = threadIdx.x; i < W; i += 256) {
        float v = r1[i];
        if (r2) v *= r2[i];
        sh[i] = v;
    }
    __syncthreads();
    const int x0 = threadIdx.x * 8;
    int lo = x0 - GF_R; if (lo < 0) lo = 0;
    int hi = x0 + GF_R; if (hi > W - 1) hi = W - 1;
    float sum = 0.f;
    for (int x = lo; x <= hi; ++x) sum += sh[x];
    float* drow = dst + (size_t)row * W;
    drow[x0] = sum;
#pragma unroll
    for (int k = 1; k < 8; ++k) {
        int xa = x0 + k + GF_R;
        int xr = x0 + k - GF_R - 1;
        if (xa < W)  sum += sh[xa];
        if (xr >= 0) sum -= sh[xr];
        drow[x0 + k] = sum;
    }
}

// vertical 121-wide box sum as a banded matmul on the WMMA pipe.
// 8 independent waves per block; each wave produces one 16x16 output tile:
//   D = sum_s A_s (16x4 band) * B_s (4x16 data)
// A layout (ISA 7.12.2, 32-bit A 16x4): V0 = {K0 | K2}, V1 = {K1 | K3}, lanes index M.
// B layout (4x16, rows striped over lane halves): V0 = {K0 | K2}, V1 = {K1 | K3}, lanes index N.
// C/D layout: VGPR j -> rows j (lanes 0-15) and j+8 (lanes 16-31).
// Image bounds are folded into the A band mask (band ∩ [0,H-1]) so B loads are
// branch-free with clamped row addresses — no per-load EXEC save/restore.
__global__ __launch_bounds__(256) void vbox_k(const float* __restrict__ src,
                                              float* __restrict__ dst) {
    const int wave = threadIdx.x >> 5;
    const int lane = threadIdx.x & 31;
    const int tx   = (blockIdx.x * 8 + wave) * 16;
    const int ty   = blockIdx.y * 16;
    const int half = lane >> 4;
    const int n    = lane & 15;
#if UDCP_WMMA
    const int m   = ty + n;                       // A-matrix row index for this lane
    const int mlo = (m - GF_R) < 0 ? 0 : (m - GF_R);
    const int mhi = (m + GF_R) > (H - 1) ? (H - 1) : (m + GF_R);
    const float* col = src + tx + n;
    v8f acc = {};
#pragma unroll 2
    for (int s = 0; s < 34; ++s) {                // 34*4 = 136 rows cover [ty-60, ty+15+60]
        const int kb = ty - GF_R + s * 4;
        const int kx = kb + (half ? 2 : 0);
        const int ky = kb + (half ? 3 : 1);
        int ckx = kx < 0 ? 0 : (kx > H - 1 ? H - 1 : kx);   // clamped (masked by A)
        int cky = ky < 0 ? 0 : (ky > H - 1 ? H - 1 : ky);
        v2f a, b;
        a.x = (kx >= mlo && kx <= mhi) ? 1.0f : 0.0f;
        a.y = (ky >= mlo && ky <= mhi) ? 1.0f : 0.0f;
        b.x = col[(size_t)ckx * W];
        b.y = col[(size_t)cky * W];
        acc = __builtin_amdgcn_wmma_f32_16x16x4_f32(
            false, a, false, b, (short)0, acc, false, false);
    }
#pragma unroll
    for (int j = 0; j < 8; ++j)
        dst[(size_t)(ty + j + half * 8) * W + tx + n] = acc[j];
#else
    // fallback: direct clamped column sums with identical output mapping
#pragma unroll
    for (int j = 0; j < 8; ++j) {
        const int y = ty + j + half * 8;
        int y0 = y - GF_R; if (y0 < 0) y0 = 0;
        int y1 = y + GF_R; if (y1 > H - 1) y1 = H - 1;
        float ssum = 0.f;
        for (int yy = y0; yy <= y1; ++yy) ssum += src[(size_t)yy * W + tx + n];
        dst[(size_t)y * W + tx + n] = ssum;
    }
#endif
}

// guided-filter coefficients: a = cov/(var+eps), b = mp - a*mI  (in-place over box buffers)
__global__ __launch_bounds__(256) void gf_ab_k(const float* bI, const float* bp,
                                               const float* bIp, const float* bII,
                                               float* a_out, float* b_out) {
    const int x = blockIdx.x * 256 + threadIdx.x;
    const int y = blockIdx.y;
    const size_t i = (size_t)y * W + x;
    const float inv_n = 1.0f / (boxcnt(x) * boxcnt(y));
    const float mI  = bI[i]  * inv_n;
    const float mp  = bp[i]  * inv_n;
    const float cov = bIp[i] * inv_n - mI * mp;
    const float var = bII[i] * inv_n - mI * mI;
    const float a = cov / (var + GFEPS);
    const float b = mp - a * mI;
    a_out[i] = a;
    b_out[i] = b;
}

// t_ref = mean(a)*I + mean(b);  J = (img - A)/max(t_ref,T0) + A, clipped
__global__ __launch_bounds__(256) void final_k(const float* __restrict__ img,
                                               const float* __restrict__ guide,
                                               const float* __restrict__ ba,
                                               const float* __restrict__ bb,
                                               const unsigned* __restrict__ am,
                                               float* __restrict__ out) {
    const int x = blockIdx.x * 256 + threadIdx.x;
    const int y = blockIdx.y;
    const size_t i = (size_t)y * W + x;
    const float inv_n = 1.0f / (boxcnt(x) * boxcnt(y));
    const float t_ref = ba[i] * inv_n * guide[i] + bb[i] * inv_n;
    const float td = fmaxf(t_ref, T0c);
#pragma unroll
    for (int c = 0; c < 3; ++c) {
        const float A = __uint_as_float(am[c]);
        float J = (img[(size_t)c * HW + i] - A) / td + A;
        out[(size_t)c * HW + i] = fminf(fmaxf(J, 0.0f), 1.0f);
    }
}

// ---------------- launcher ----------------
// ws usage: 5 fp32 planes (83.9 MB) + 8208 u32 stats. d_out planes used as scratch
// (bI/bp/bIp then a/b) before final_k fully rewrites them. Everything read is fully
// written earlier in the same call -> deterministic, no cross-call state.
extern "C" void kernel_launch(void* const* d_in, const int* in_sizes, int n_in,
                              void* d_out, int out_size, void* d_ws, size_t ws_size,
                              hipStream_t stream) {
    (void)in_sizes; (void)n_in; (void)out_size; (void)ws_size;
    const float* img = (const float*)d_in[0];
    float* out = (float*)d_out;

    float* P0 = (float*)d_ws;      // guide I
    float* P1 = P0 + HW;           // mgb -> rt -> t
    float* P2 = P1 + HW;           // separable-pass scratch
    float* P3 = P2 + HW;           // dc -> bII -> box(a)
    float* P4 = P3 + HW;           // box(b)
    unsigned* stats   = (unsigned*)(P4 + HW);
    unsigned* hist_hi = stats;              // 4096
    unsigned* hist_lo = stats + 4096;       // 4096
    unsigned* sel     = stats + 8192;       // [0]=hi bin [1]=remaining [2]=thr bits
    unsigned* am      = stats + 8195;       // A as fp32 bit-max, r/g/b

    float* D0 = out;               // bI -> b -> J_r
    float* D1 = out + HW;          // bp      -> J_g
    float* D2 = out + 2 * HW;      // bIp -> a -> J_b

    dim3 b256(256), b32(32);
    dim3 gElem(HW / 256);
    dim3 gRow(W / 256, H);
    dim3 gTile(W / (16 * 8), H / 16);   // 8 waves/block, one 16x16 tile per wave

    zero_stats_k<<<dim3((8208 + 255) / 256), b256, 0, stream>>>(stats);
    prep_k<<<gElem, b256, 0, stream>>>(img, P0, P1);

    // dark channel
    hmin_k<<<gRow, b256, 0, stream>>>(P1, P2);
    vmin_k<<<gRow, b256, 0, stream>>>(P2, P3, 0);

    // atmospheric light A via 24-bit float-radix top-k threshold
    hist_hi_k<<<gElem, b256, 0, stream>>>(P3, hist_hi);
    scan_k<<<dim3(1), b32, 0, stream>>>(hist_hi, sel, 0);
    hist_lo_k<<<gElem, b256, 0, stream>>>(P3, sel, hist_lo);
    scan_k<<<dim3(1), b32, 0, stream>>>(hist_lo, sel, 1);
    amax_k<<<gElem, b256, 0, stream>>>(img, P3, sel, am);

    // transmission estimate
    tprep_k<<<gElem, b256, 0, stream>>>(img, am, P1);
    hmin_k<<<gRow, b256, 0, stream>>>(P1, P2);
    vmin_k<<<gRow, b256, 0, stream>>>(P2, P1, 1);   // t in P1

    // guided filter: box(I), box(p), box(I*p), box(I*I)
    hbox_k<<<dim3(H), b256, 0, stream>>>(P0, (const float*)nullptr, P2);
    vbox_k<<<gTile, b256, 0, stream>>>(P2, D0);
    hbox_k<<<dim3(H), b256, 0, stream>>>(P1, (const float*)nullptr, P2);
    vbox_k<<<gTile, b256, 0, stream>>>(P2, D1);
    hbox_k<<<dim3(H), b256, 0, stream>>>(P0, P1, P2);
    vbox_k<<<gTile, b256, 0, stream>>>(P2, D2);
    hbox_k<<<dim3(H), b256, 0, stream>>>(P0, P0, P2);
    vbox_k<<<gTile, b256, 0, stream>>>(P2, P3);

    gf_ab_k<<<gRow, b256, 0, stream>>>(D0, D1, D2, P3, D2, D0);  // a->D2, b->D0

    hbox_k<<<dim3(H), b256, 0, stream>>>(D2, (const float*)nullptr, P2);
    vbox_k<<<gTile, b256, 0, stream>>>(P2, P3);                  // box(a)
    hbox_k<<<dim3(H), b256, 0, stream>>>(D0, (const float*)nullptr, P2);
    vbox_k<<<gTile, b256, 0, stream>>>(P2, P4);                  // box(b)

    final_k<<<gRow, b256, 0, stream>>>(img, P0, P3, P4, am, out);
}